// NodeNetGNN_44925357916689
// MI455X (gfx1250) — compile-verified
//
#include <hip/hip_runtime.h>
#include <math.h>

// ---------------------------------------------------------------------------
// NodeNetGNN for MI455X (gfx1250, wave32, WMMA).
//
// Dense math runs on V_WMMA_F32_16X16X4_F32 (exact fp32; total dense work is
// only ~7.5 GFLOP so low-precision WMMA buys nothing — this problem is
// scatter/latency bound, not compute bound).  NNConv is reformulated as
//   M = Z (E x 544) @ [W_topo ; b_topo] (544 x 32),
//   Z[e, p*32+i] = concat(ef_e,1)[p] * h_src[e][i]
// with Z fragments built on the fly from LDS-staged gathers (the E x 32 x 32
// per-edge weight tensor is never materialized).  The 68KB B matrix is staged
// into LDS once per block via GLOBAL_LOAD_ASYNC_TO_LDS_B128 (ASYNCcnt),
// overlapped with the per-tile h/ef gathers.  Per-destination norms
// (deg^-1/2 for GraphConv, 1/deg for NNConv) are folded into the edge
// contributions so each output tensor gets exactly one atomic-accumulate pass.
// ---------------------------------------------------------------------------

typedef float v2f __attribute__((ext_vector_type(2)));
typedef float v4f __attribute__((ext_vector_type(4)));
typedef float v8f __attribute__((ext_vector_type(8)));
typedef int   v4i __attribute__((ext_vector_type(4)));

#define WMMA_F32(a, b, c) \
  __builtin_amdgcn_wmma_f32_16x16x4_f32(false, (a), false, (b), (short)0, (c), false, false)

#if __has_builtin(__builtin_amdgcn_global_load_async_to_lds_b128) && \
    __has_builtin(__builtin_amdgcn_s_wait_asynccnt)
#define GNN_ASYNC_LDS 1
#else
#define GNN_ASYNC_LDS 0
#endif

__device__ __forceinline__ void atomAddF(float* p, float v) {
  __hip_atomic_fetch_add(p, v, __ATOMIC_RELAXED, __HIP_MEMORY_SCOPE_AGENT);
}

// ------------------------------ utility kernels ----------------------------

__global__ void gnn_zero(float* __restrict__ p, int n) {
  int t = blockIdx.x * blockDim.x + threadIdx.x;
  if (t < n) p[t] = 0.0f;
}

__global__ void gnn_count(const int* __restrict__ idx, float* __restrict__ deg, int n) {
  int t = blockIdx.x * blockDim.x + threadIdx.x;
  if (t < n) atomAddF(&deg[idx[t]], 1.0f);
}

// mode 0: deg -> deg^-1/2 (GraphConv norm), mode 1: deg -> 1/deg (NNConv mean)
__global__ void gnn_xform(float* __restrict__ a, int n, int mode) {
  int t = blockIdx.x * blockDim.x + threadIdx.x;
  if (t < n) {
    float d = fmaxf(a[t], 1.0f);
    a[t] = (mode == 0) ? (1.0f / sqrtf(d)) : (1.0f / d);
  }
}

// --------------------- dense  Y = (X * s[:,None]) @ W(32x32) ----------------
// One wave per 16-row tile; two 16x16 fp32 WMMA accumulators (N=0..15,16..31).

__global__ __launch_bounds__(256) void gnn_gemm32(
    const float* __restrict__ X, const float* __restrict__ scale,
    const float* __restrict__ W, float* __restrict__ Y, int nrows) {
  const int lane = threadIdx.x & 31;
  const int wid  = threadIdx.x >> 5;
  const int tile = blockIdx.x * 8 + wid;
  const int m0   = tile * 16;
  if (m0 >= nrows) return;  // whole-wave uniform exit

  const int nlo  = lane & 15;
  const int koff = (lane >> 4) << 1;          // A/B K sub-offset per half-wave
  int row = m0 + nlo;
  if (row >= nrows) row = nrows - 1;          // clamp (stores are guarded)
  const float s = scale ? scale[row] : 1.0f;
  const float* xr = X + (size_t)row * 32;

  v8f c0 = {0.f,0.f,0.f,0.f,0.f,0.f,0.f,0.f};
  v8f c1 = c0;
#pragma unroll
  for (int k = 0; k < 32; k += 4) {
    v2f a, b0, b1;
    a.x = xr[k + koff] * s;
    a.y = xr[k + koff + 1] * s;
    const float* w0 = W + (k + koff) * 32;
    const float* w1 = w0 + 32;
    b0.x = w0[nlo];      b0.y = w1[nlo];
    b1.x = w0[nlo + 16]; b1.y = w1[nlo + 16];
    c0 = WMMA_F32(a, b0, c0);
    c1 = WMMA_F32(a, b1, c1);
  }
  const int mh = m0 + ((lane >> 4) << 3);
#pragma unroll
  for (int r = 0; r < 8; ++r) {
    int m = mh + r;
    if (m < nrows) {
      float* yr = Y + (size_t)m * 32;
      yr[nlo]      = c0[r];
      yr[nlo + 16] = c1[r];
    }
  }
}

// -------- GraphConv scatter: out[dst] += feat[src] * dscale[dst] -----------

__global__ void gnn_scatter(const float* __restrict__ feat, const int* __restrict__ src,
                            const int* __restrict__ dst, const float* __restrict__ dscale,
                            float* __restrict__ out, int nedge) {
  int t = blockIdx.x * blockDim.x + threadIdx.x;
  if (t >= nedge * 32) return;
  int e = t >> 5, c = t & 31;
  int sI = src[e], d = dst[e];
  atomAddF(&out[(size_t)d * 32 + c], feat[(size_t)sI * 32 + c] * dscale[d]);
}

// ---------------------------- fused NNConv ---------------------------------
// Per 16-edge tile (one wave): gather h_src rows + edge feats into LDS, run
// K=544 WMMA GEMM against block-staged [W_topo ; b_topo], scatter with 1/deg.

__global__ __launch_bounds__(256) void gnn_nnconv(
    const float* __restrict__ Hsrc, const float* __restrict__ EF,
    const float* __restrict__ Wt, const float* __restrict__ bt,
    const int* __restrict__ src, const int* __restrict__ dst,
    const float* __restrict__ invdeg, float* __restrict__ out,
    int nedge, int ntiles) {
  __shared__ float B_s[544 * 32];   // [W_topo ; b_topo], row-major, 68KB
  __shared__ float h_s[8][16][33];  // padded: lanes 0..15 read distinct banks
  __shared__ float g_s[8][16][17];

  const int tid  = threadIdx.x;
  const int lane = tid & 31;
  const int wid  = tid >> 5;
  const int tile = blockIdx.x * 8 + wid;
  const bool active = (tile < ntiles);
  const int m0 = active ? tile * 16 : 0;

  // ---- stage B into LDS (async DMA, overlapped with the gathers below) ----
#if GNN_ASYNC_LDS
  for (int i = tid; i < (512 * 32) / 4; i += 256)
    __builtin_amdgcn_global_load_async_to_lds_b128(
        (v4i*)(Wt + i * 4), (v4i*)(&B_s[i * 4]), 0, 0);
  for (int i = tid; i < (32 * 32) / 4; i += 256)
    __builtin_amdgcn_global_load_async_to_lds_b128(
        (v4i*)(bt + i * 4), (v4i*)(&B_s[512 * 32 + i * 4]), 0, 0);
#else
  for (int i = tid; i < (512 * 32) / 4; i += 256)
    ((v4f*)B_s)[i] = ((const v4f*)Wt)[i];
  for (int i = tid; i < (32 * 32) / 4; i += 256)
    ((v4f*)B_s)[(512 * 32) / 4 + i] = ((const v4f*)bt)[i];
#endif

  // ---- gather this tile's h rows and edge feats into LDS ----
  if (active) {
#pragma unroll 4
    for (int r = 0; r < 16; ++r) {
      int e = m0 + r;
      float mask = (e < nedge) ? 1.0f : 0.0f;
      int ec = (e < nedge) ? e : (nedge - 1);
      int sidx = src[ec];
      h_s[wid][r][lane] = Hsrc[(size_t)sidx * 32 + lane] * mask;
      if (lane < 16) g_s[wid][r][lane] = EF[(size_t)ec * 16 + lane] * mask;
    }
  }
#if GNN_ASYNC_LDS
  __builtin_amdgcn_s_wait_asynccnt(0);
#endif
  __syncthreads();
  if (!active) return;

  const int nlo  = lane & 15;
  const int koff = (lane >> 4) << 1;
  const float* hrow = &h_s[wid][lane & 15][0];
  const float* grow = &g_s[wid][lane & 15][0];

  v8f c0 = {0.f,0.f,0.f,0.f,0.f,0.f,0.f,0.f};
  v8f c1 = c0;

  // K = 0..511 : A[k] = g[k>>5] * h[k&31],  B row k = B_s + k*32
#pragma unroll 4
  for (int k = 0; k < 512; k += 4) {
    int kk = k + koff;
    v2f a, b0, b1;
    a.x = grow[kk >> 5] * hrow[kk & 31];
    a.y = grow[(kk + 1) >> 5] * hrow[(kk + 1) & 31];
    const float* w0 = &B_s[kk * 32];
    b0.x = w0[nlo];      b0.y = w0[32 + nlo];
    b1.x = w0[nlo + 16]; b1.y = w0[32 + nlo + 16];
    c0 = WMMA_F32(a, b0, c0);
    c1 = WMMA_F32(a, b1, c1);
  }
  // K = 512..543 : A = h (implicit g=1 row), B rows are the staged b_topo
#pragma unroll
  for (int k = 512; k < 544; k += 4) {
    int kk = k + koff;
    v2f a, b0, b1;
    a.x = hrow[kk - 512];
    a.y = hrow[kk - 511];
    const float* w0 = &B_s[kk * 32];
    b0.x = w0[nlo];      b0.y = w0[32 + nlo];
    b1.x = w0[nlo + 16]; b1.y = w0[32 + nlo + 16];
    c0 = WMMA_F32(a, b0, c0);
    c1 = WMMA_F32(a, b1, c1);
  }

  const int mh = m0 + ((lane >> 4) << 3);
#pragma unroll
  for (int r = 0; r < 8; ++r) {
    int e = mh + r;
    int ec = (e < nedge) ? e : (nedge - 1);   // padded rows carry exact zeros
    int d = dst[ec];
    float w = invdeg[d];
    atomAddF(&out[(size_t)d * 32 + nlo],      c0[r] * w);
    atomAddF(&out[(size_t)d * 32 + nlo + 16], c1[r] * w);
  }
}

// ----------------------- final bias / residual add ------------------------

__global__ void gnn_finalize(float* __restrict__ out, const float* __restrict__ extra,
                             const float* __restrict__ bA, const float* __restrict__ bB,
                             int n) {
  int t = blockIdx.x * blockDim.x + threadIdx.x;
  if (t >= n) return;
  int c = t & 31;
  float v = out[t] + bA[c] + bB[c];
  if (extra) v += extra[t];
  out[t] = v;
}

// ---------------------------------------------------------------------------

extern "C" void kernel_launch(void* const* d_in, const int* in_sizes, int n_in,
                              void* d_out, int out_size, void* d_ws, size_t ws_size,
                              hipStream_t stream) {
  const float* node_feat   = (const float*)d_in[0];
  const float* net_feat    = (const float*)d_in[1];
  const float* pin_feat    = (const float*)d_in[2];
  const float* hanna_feat  = (const float*)d_in[3];
  const float* edge_feat   = (const float*)d_in[4];
  const int* pins_src      = (const int*)d_in[5];
  const int* pins_dst      = (const int*)d_in[6];
  const int* pinned_src    = (const int*)d_in[7];
  const int* pinned_dst    = (const int*)d_in[8];
  const int* connect_src   = (const int*)d_in[9];
  const int* connect_dst   = (const int*)d_in[10];
  const int* pt_src        = (const int*)d_in[11];
  const int* pt_dst        = (const int*)d_in[12];
  const int* pf_src        = (const int*)d_in[13];
  const int* pf_dst        = (const int*)d_in[14];
  const float* W_net       = (const float*)d_in[15];
  const float* b_net       = (const float*)d_in[16];
  const float* W_topo      = (const float*)d_in[17];
  const float* b_topo      = (const float*)d_in[18];
  const float* W_pins      = (const float*)d_in[19];
  const float* b_pins      = (const float*)d_in[20];
  const float* W_connect   = (const float*)d_in[21];
  const float* b_connect   = (const float*)d_in[22];
  const float* W_pt        = (const float*)d_in[23];
  const float* b_pt        = (const float*)d_in[24];
  const float* b_pinned    = (const float*)d_in[25];
  const float* b_pf        = (const float*)d_in[26];

  const int nCell = in_sizes[0] / 32;
  const int nNet  = in_sizes[1] / 32;
  const int nG    = in_sizes[3] / 32;
  const int E_    = in_sizes[5];

  // workspace layout (floats)
  float* ws = (float*)d_ws;
  float* s_pins_src   = ws; ws += nCell;   // deg^-1/2
  float* s_pins_dst   = ws; ws += nNet;    // deg^-1/2
  float* s_pinned_dst = ws; ws += nCell;   // 1/deg
  float* s_pf_dst     = ws; ws += nCell;   // 1/deg
  float* s_conn_src   = ws; ws += nG;      // deg^-1/2
  float* s_conn_dst   = ws; ws += nG;      // deg^-1/2
  float* s_pt_src     = ws; ws += nCell;   // deg^-1/2
  float* s_pt_dst     = ws; ws += nG;      // deg^-1/2
  const int degTotal = 4 * nCell + nNet + 3 * nG;
  float* feat_pins = ws; ws += (size_t)nCell * 32;
  float* feat_conn = ws; ws += (size_t)nG * 32;
  float* feat_pt   = ws; ws += (size_t)nCell * 32;
  float* feat_net  = ws; ws += (size_t)nNet * 32;

  float* out_cell  = (float*)d_out;
  float* out_net   = out_cell + (size_t)nCell * 32;
  float* out_gcell = out_net + (size_t)nNet * 32;

  auto cdiv = [](int a, int b) { return (a + b - 1) / b; };

  // 0) zero accumulators + degree counters
  gnn_zero<<<cdiv(out_size, 256), 256, 0, stream>>>((float*)d_out, out_size);
  gnn_zero<<<cdiv(degTotal, 256), 256, 0, stream>>>(s_pins_src, degTotal);

  // 1) degree counts
  const int gE = cdiv(E_, 256);
  gnn_count<<<gE, 256, 0, stream>>>(pins_src,    s_pins_src,   E_);
  gnn_count<<<gE, 256, 0, stream>>>(pins_dst,    s_pins_dst,   E_);
  gnn_count<<<gE, 256, 0, stream>>>(pinned_dst,  s_pinned_dst, E_);
  gnn_count<<<gE, 256, 0, stream>>>(pf_dst,      s_pf_dst,     E_);
  gnn_count<<<gE, 256, 0, stream>>>(connect_src, s_conn_src,   E_);
  gnn_count<<<gE, 256, 0, stream>>>(connect_dst, s_conn_dst,   E_);
  gnn_count<<<gE, 256, 0, stream>>>(pt_src,      s_pt_src,     E_);
  gnn_count<<<gE, 256, 0, stream>>>(pt_dst,      s_pt_dst,     E_);

  // 2) degree -> normalization scalars
  gnn_xform<<<cdiv(nCell, 256), 256, 0, stream>>>(s_pins_src,   nCell, 0);
  gnn_xform<<<cdiv(nNet,  256), 256, 0, stream>>>(s_pins_dst,   nNet,  0);
  gnn_xform<<<cdiv(nCell, 256), 256, 0, stream>>>(s_pinned_dst, nCell, 1);
  gnn_xform<<<cdiv(nCell, 256), 256, 0, stream>>>(s_pf_dst,     nCell, 1);
  gnn_xform<<<cdiv(nG,    256), 256, 0, stream>>>(s_conn_src,   nG,    0);
  gnn_xform<<<cdiv(nG,    256), 256, 0, stream>>>(s_conn_dst,   nG,    0);
  gnn_xform<<<cdiv(nCell, 256), 256, 0, stream>>>(s_pt_src,     nCell, 0);
  gnn_xform<<<cdiv(nG,    256), 256, 0, stream>>>(s_pt_dst,     nG,    0);

  // 3) dense WMMA GEMMs: feat = (X * deg_src^-1/2) @ W
  auto gemmGrid = [&](int rows) { return cdiv(cdiv(rows, 16), 8); };
  gnn_gemm32<<<gemmGrid(nCell), 256, 0, stream>>>(node_feat,  s_pins_src, W_pins,    feat_pins, nCell);
  gnn_gemm32<<<gemmGrid(nG),    256, 0, stream>>>(hanna_feat, s_conn_src, W_connect, feat_conn, nG);
  gnn_gemm32<<<gemmGrid(nCell), 256, 0, stream>>>(node_feat,  s_pt_src,   W_pt,      feat_pt,   nCell);
  gnn_gemm32<<<gemmGrid(nNet),  256, 0, stream>>>(net_feat,   nullptr,    W_net,     feat_net,  nNet);

  // 4) GraphConv scatters (dst norm folded in)
  const int gS = cdiv(E_ * 32, 256);
  gnn_scatter<<<gS, 256, 0, stream>>>(feat_pins, pins_src,    pins_dst,    s_pins_dst, out_net,   E_);
  gnn_scatter<<<gS, 256, 0, stream>>>(feat_conn, connect_src, connect_dst, s_conn_dst, out_gcell, E_);
  gnn_scatter<<<gS, 256, 0, stream>>>(feat_pt,   pt_src,      pt_dst,      s_pt_dst,   out_gcell, E_);

  // 5) fused NNConvs (async-staged B + gather + WMMA K=544 + scatter w/ 1/deg)
  const int ntiles = cdiv(E_, 16);
  const int gNN = cdiv(ntiles, 8);
  gnn_nnconv<<<gNN, 256, 0, stream>>>(net_feat,   pin_feat,  W_topo, b_topo,
                                      pinned_src, pinned_dst, s_pinned_dst, out_cell, E_, ntiles);
  gnn_nnconv<<<gNN, 256, 0, stream>>>(hanna_feat, edge_feat, W_topo, b_topo,
                                      pf_src,     pf_dst,     s_pf_dst,     out_cell, E_, ntiles);

  // 6) biases + residual GEMM term
  gnn_finalize<<<cdiv(nCell * 32, 256), 256, 0, stream>>>(out_cell,  nullptr,  b_pinned,  b_pf,   nCell * 32);
  gnn_finalize<<<cdiv(nNet  * 32, 256), 256, 0, stream>>>(out_net,   feat_net, b_pins,    b_net,  nNet * 32);
  gnn_finalize<<<cdiv(nG    * 32, 256), 256, 0, stream>>>(out_gcell, nullptr,  b_connect, b_pt,   nG * 32);
}